// ANFISNet_81114752352952
// MI455X (gfx1250) — compile-verified
//
#include <hip/hip_runtime.h>
#include <hip/hip_bf16.h>
#include <stdint.h>

// CDNA5 WMMA fragment types: f32 16x16x4 -> A/B = 2 VGPRs (v2f), C/D = 8 VGPRs (v8f)
typedef __attribute__((ext_vector_type(2))) float v2f;
typedef __attribute__((ext_vector_type(8))) float v8f;

// 12-bit antecedent masks for the 30 rules (bit k set <=> membership k is an antecedent).
// Rows 30,31 are zero padding for the second 16-rule WMMA tile.
#define RM(a,b,cc,d) (uint16_t)((1u<<(a))|(1u<<(b))|(1u<<(cc))|(1u<<(d)))
__device__ __constant__ uint16_t RULE_MASK[32] = {
    RM(0,3,8,9),  RM(0,3,8,10), RM(0,3,7,9),  RM(0,4,8,9),
    RM(0,3,7,10), RM(0,4,8,10), RM(1,3,8,9),  RM(0,5,8,9),
    RM(1,3,8,10), RM(0,3,6,9),  RM(0,4,7,10), RM(1,4,7,9),
    RM(1,3,7,10), RM(1,3,6,9),  RM(0,5,7,10), RM(0,4,6,9),
    RM(1,4,8,10), RM(1,3,7,9),  RM(2,3,7,9),  RM(0,5,6,9),
    RM(1,5,7,9),  RM(2,4,8,9),  RM(2,5,6,10), RM(2,5,6,9),
    RM(2,4,6,10), RM(1,5,6,10), RM(2,5,7,10), RM(1,5,6,9),
    RM(2,4,6,9),  RM(1,4,6,10), 0, 0
};
#undef RM

// ANFIS forward, log2-space formulation:
//   q_k = -((x[dim(k)]-c_k) * g_k)^2, g_k = sqrt(0.5*log2 e)/sigma_k  (11 values, K padded to 12)
//   log2w = A(30x12, binary) @ q      <-- V_WMMA_F32_16X16X4_F32, 2 M-tiles x 3 K-steps
//   w = exp2(log2w)  (native v_exp_f32, no ln->log2 scaling needed)
//   out = (w . clip(conseq,0,100)) / max(sum w, 1e-8)
__global__ __launch_bounds__(256) void anfis_wmma_kernel(
    const float* __restrict__ x,       // (batch, 4)
    const float* __restrict__ c,       // (11,)
    const float* __restrict__ log_s,   // (11,)
    const float* __restrict__ conseq,  // (30,)
    float* __restrict__ out,           // (batch, 1)
    int batch)
{
    const int lane = threadIdx.x & 31;        // wave32
    const int hi   = lane >> 4;               // half-wave: selects K pair / rule half
    const int col  = lane & 15;               // tile column (batch row) and A-matrix row

    const int waveInBlock   = threadIdx.x >> 5;
    const int wavesPerBlock = blockDim.x >> 5;
    const int globalWave    = blockIdx.x * wavesPerBlock + waveInBlock;
    const int numWaves      = gridDim.x * wavesPerBlock;
    const int numTiles      = batch >> 4;     // 16 rows per tile

    // sqrt(0.5 * log2(e)) : folds both the -0.5 and the ln->log2 conversion into invK,
    // so per-element work is just (x-c)*g then a negated square, and exp2 needs no scaling.
    const float SQRT_HALF_LOG2E = 0.84932180028801904f;

    // ---------------- per-wave constant setup (amortized over the tile loop) -------------
    // B-operand k indices owned by this lane: kk[j] = 4*(j>>1) + 2*hi + (j&1), j = 0..5
    float cK[6], invK[6]; int dimK[6];
#pragma unroll
    for (int j = 0; j < 6; ++j) {
        const int k = 4 * (j >> 1) + 2 * hi + (j & 1);
        if (k < 11) {
            cK[j]   = c[k];
            float s = fmaxf(__expf(log_s[k]), 0.001f);
            invK[j] = SQRT_HALF_LOG2E / s;
            dimK[j] = k / 3;                  // DIM_MAP = [0,0,0,1,1,1,2,2,2,3,3]
        } else {                              // K=11 zero padding
            cK[j] = 0.0f; invK[j] = 0.0f; dimK[j] = 0;
        }
    }

    // A fragments (constant 0/1 selection matrix):
    //   row M = 16*m + (lane&15); lanes 0-15 hold K pair (4t+0,4t+1), lanes 16-31 (4t+2,4t+3)
    v2f afrag[2][3];
#pragma unroll
    for (int m = 0; m < 2; ++m) {
        const uint32_t mask = RULE_MASK[16 * m + col];
#pragma unroll
        for (int t = 0; t < 3; ++t) {
            const int kb = 4 * t + 2 * hi;
            afrag[m][t].x = ((mask >> (kb    )) & 1u) ? 1.0f : 0.0f;
            afrag[m][t].y = ((mask >> (kb + 1)) & 1u) ? 1.0f : 0.0f;
        }
    }

    // Clipped consequents for this lane's D rows: rule(m,j) = 16*m + 8*hi + j
    float cq[2][8];
#pragma unroll
    for (int m = 0; m < 2; ++m)
#pragma unroll
        for (int j = 0; j < 8; ++j) {
            const int r = 16 * m + 8 * hi + j;
            cq[m][j] = (r < 30) ? fminf(fmaxf(conseq[r], 0.0f), 100.0f) : 0.0f;
        }

    // Kill padded rules 30,31 (tile1, VGPRs 6,7, upper half-wave): C = -1e30 -> exp2 -> 0.
    const float padC = hi ? -1e30f : 0.0f;

    // ---------------- main loop: one 16-row tile per wave iteration ----------------------
    for (int tile = globalWave; tile < numTiles; tile += numWaves) {
        const int row = (tile << 4) + col;
        const float4 xv = *(const float4*)(x + 4 * row);   // 128b coalesced load

        // q_k (in log2 units) for this lane's 6 K indices; negate folds into mul modifier
        float q[6];
#pragma unroll
        for (int j = 0; j < 6; ++j) {
            const int d  = dimK[j];
            const float xd = (d == 0) ? xv.x : (d == 1) ? xv.y : (d == 2) ? xv.z : xv.w;
            const float z  = (xd - cK[j]) * invK[j];
            q[j] = -(z * z);
        }

        // B fragments: column N = lane&15, K pairs matching the A split
        v2f bfrag[3];
#pragma unroll
        for (int t = 0; t < 3; ++t) { bfrag[t].x = q[2 * t]; bfrag[t].y = q[2 * t + 1]; }

        v8f acc0 = {};
        v8f acc1 = {};
        acc1[6] = padC;
        acc1[7] = padC;

        // log2w = A @ q : 2 rule tiles x 3 K-steps of V_WMMA_F32_16X16X4_F32
#pragma unroll
        for (int t = 0; t < 3; ++t) {
            acc0 = __builtin_amdgcn_wmma_f32_16x16x4_f32(
                false, afrag[0][t], false, bfrag[t], (short)0, acc0, false, false);
            acc1 = __builtin_amdgcn_wmma_f32_16x16x4_f32(
                false, afrag[1][t], false, bfrag[t], (short)0, acc1, false, false);
        }

        // w = exp2(log2w) via native v_exp_f32; partial sums over this lane's 16 rules
        float ssum = 0.0f, tsum = 0.0f;
#pragma unroll
        for (int j = 0; j < 8; ++j) {
            const float w0 = __builtin_amdgcn_exp2f(acc0[j]);
            const float w1 = __builtin_amdgcn_exp2f(acc1[j]);
            ssum += w0 + w1;
            tsum += w0 * cq[0][j] + w1 * cq[1][j];
        }

        // lanes l and l+16 hold complementary rule halves of the same column
        ssum += __shfl_xor(ssum, 16, 32);
        tsum += __shfl_xor(tsum, 16, 32);

        const float res = tsum / fmaxf(ssum, 1e-8f);
        if (!hi) out[row] = res;   // 16 contiguous stores per wave
    }
}

extern "C" void kernel_launch(void* const* d_in, const int* in_sizes, int n_in,
                              void* d_out, int out_size, void* d_ws, size_t ws_size,
                              hipStream_t stream) {
    const float* x      = (const float*)d_in[0];
    const float* c      = (const float*)d_in[1];
    const float* log_s  = (const float*)d_in[2];
    const float* conseq = (const float*)d_in[3];
    float* out          = (float*)d_out;

    const int batch    = in_sizes[0] / 4;   // x is (batch, 4)
    const int numTiles = batch >> 4;

    // Persistent waves: 2048 blocks x 8 wave32 = 16384 waves; grid-stride over tiles.
    int blocks = 2048;
    int maxBlocks = (numTiles + 7) / 8;     // at least one tile per wave if batch is small
    if (blocks > maxBlocks) blocks = (maxBlocks > 0) ? maxBlocks : 1;

    anfis_wmma_kernel<<<blocks, 256, 0, stream>>>(x, c, log_s, conseq, out, batch);
}